// SSIMLoss_66795331388107
// MI455X (gfx1250) — compile-verified
//
#include <hip/hip_runtime.h>
#include <math.h>

// ---------------------------------------------------------------------------
// SSIM loss (softmax + one-hot + 11x11 separable Gaussian SSIM) for MI455X.
// Gaussian conv passes are band-matrix matmuls on V_WMMA_F32_16X16X4_F32:
// a 16-wide output group consumes a 26-wide input window as 7 chained K=4
// WMMAs; band entries outside [0,10] are zero, killing the overhang.
// All per-wave job selection is readfirstlane-uniform so WMMA runs with
// EXEC all-ones and the 7-WMMA chains issue back-to-back.
// ---------------------------------------------------------------------------

typedef __attribute__((ext_vector_type(2))) float v2f;
typedef __attribute__((ext_vector_type(8))) float v8f;

#define BB 8
#define CC 19
#define HH 512
#define WW 512
#define HWc (HH * WW)
#define OUTD 502            // 512 - 11 + 1
#define NIMG (BB * CC)      // 152

// Normalized 11-tap Gaussian, sigma = 1.5
__constant__ float G11[11] = {
    0.00102838f, 0.00759876f, 0.03600077f, 0.10936069f, 0.21300553f,
    0.26601172f, 0.21300553f, 0.10936069f, 0.03600077f, 0.00759876f,
    0.00102838f};

__device__ __forceinline__ float gval(int j) {
  return ((unsigned)j <= 10u) ? G11[j] : 0.0f;
}

// ---------------------------------------------------------------------------
// Kernel 0: zero the per-(b,c) accumulators (ws is poisoned, not re-zeroed)
// ---------------------------------------------------------------------------
__global__ void zero_k(float* __restrict__ acc) {
  int t = threadIdx.x;
  if (t < NIMG) acc[t] = 0.0f;
}

// ---------------------------------------------------------------------------
// Kernel 1: channel softmax  pred[B,C,H,W] -> p[B,C,H,W]  (one thread/pixel)
// ---------------------------------------------------------------------------
__global__ __launch_bounds__(256) void softmax_k(const float* __restrict__ pred,
                                                 float* __restrict__ p) {
  int pi = blockIdx.x * blockDim.x + threadIdx.x;
  if (pi >= BB * HWc) return;
  int b = pi / HWc;
  int rem = pi - b * HWc;
  const float* base = pred + (size_t)b * CC * HWc + rem;

  float v[CC];
  float m = -3.4e38f;
#pragma unroll
  for (int c = 0; c < CC; ++c) {
    v[c] = base[c * HWc];
    m = fmaxf(m, v[c]);
  }
  float s = 0.0f;
#pragma unroll
  for (int c = 0; c < CC; ++c) {
    v[c] = __expf(v[c] - m);
    s += v[c];
  }
  float inv = 1.0f / s;
  float* ob = p + (size_t)b * CC * HWc + rem;
#pragma unroll
  for (int c = 0; c < CC; ++c) ob[c * HWc] = v[c] * inv;
}

// ---------------------------------------------------------------------------
// Kernel 2: fused separable Gaussian convs (WMMA f32) + SSIM + accumulation.
// One block = one 32x32 output tile of one (b,c) image.
//   signals: s0 = p -> mu1 ; s1 = p*p -> E[pp] ; s2 = p*t -> E[pt] ;
//            s3 = t -> mu2  (t one-hot => conv(t^2) == mu2)
// ---------------------------------------------------------------------------
__global__ __launch_bounds__(256) void ssim_conv_k(const float* __restrict__ P,
                                                   const int* __restrict__ T,
                                                   float* __restrict__ acc) {
  __shared__ __align__(16) float sIn[4][48][44];  // 4 signals, clamped borders
  __shared__ __align__(16) float sHT[4][32][50];  // hconv, TRANSPOSED [col][row]
  __shared__ __align__(16) float sO[4][32][36];   // final conv values
  __shared__ float sRed[256];

  const int tile = blockIdx.x;      // 0..255 (16x16 tiles of 32x32 outputs)
  const int img  = blockIdx.y;      // 0..151
  const int b = img / CC, c = img - b * CC;
  const int tr0 = (tile >> 4) * 32;
  const int tc0 = (tile & 15) * 32;
  const int tid = threadIdx.x;

  // ---- load p + one-hot mask, build all 4 signals in LDS (clamped borders:
  //      halo garbage is finite and annihilated by zero band entries) ----
  const float* Pp = P + (size_t)img * HWc;
  const int*   Tp = T + (size_t)b * HWc;
  for (int i = tid; i < 48 * 44; i += 256) {
    int r = i / 44, q = i - (i / 44) * 44;
    int gr = min(tr0 + r, HH - 1);
    int gc = min(tc0 + q, WW - 1);
    float pv = Pp[gr * WW + gc];
    float mv = (Tp[gr * WW + gc] == c) ? 1.0f : 0.0f;
    sIn[0][r][q] = pv;
    sIn[1][r][q] = pv * pv;
    sIn[2][r][q] = pv * mv;
    sIn[3][r][q] = mv;
  }

  const int lane = tid & 31;
  const int lmod = lane & 15;   // M (A) / N (B,C/D) index
  const int lhi  = lane >> 4;   // K-pair selector
  // wave id as an SGPR -> all job decomposition is scalar control flow
  const int wu = __builtin_amdgcn_readfirstlane(tid >> 5);

  // Band operand pairs (identical for horizontal-B and vertical-A):
  // chunk q covers window cols j0..j0+1 (per lane); value g[j - n], n = lmod.
  v2f band[7];
#pragma unroll
  for (int q = 0; q < 7; ++q) {
    const int j0 = 4 * q + 2 * lhi;
    band[q].x = gval(j0 - lmod);
    band[q].y = gval(j0 + 1 - lmod);
  }

  __syncthreads();

  // ---- horizontal pass: h(r, n) = sum_j In(r, c0+j) * g[j-n] ----
  // 24 jobs = 4 signals x 3 rowgroups(48 rows) x 2 colgroups(32 cols)
#pragma unroll
  for (int rep = 0; rep < 3; ++rep) {
    const int job = wu + rep * 8;
    const int s = job & 3;
    const int rc = job >> 2;
    const int r0 = (rc >> 1) * 16;   // 0,16,32
    const int c0 = (rc & 1) * 16;    // 0,16
    v8f d = {0.f, 0.f, 0.f, 0.f, 0.f, 0.f, 0.f, 0.f};
#pragma unroll
    for (int q = 0; q < 7; ++q) {
      const int j0 = 4 * q + 2 * lhi;
      v2f a = *(const v2f*)&sIn[s][r0 + lmod][c0 + j0];   // ds_load_b64
      d = __builtin_amdgcn_wmma_f32_16x16x4_f32(false, a, false, band[q],
                                                (short)0, d, false, false);
    }
    // D[m][n]: lane holds (M = v + 8*lhi, N = lmod); store transposed
#pragma unroll
    for (int v = 0; v < 8; ++v)
      sHT[s][c0 + lmod][r0 + 8 * lhi + v] = d[v];
  }
  __syncthreads();

  // ---- vertical pass: out(m, n) = sum_j g[j-m] * h(r0+j, c0+n) ----
  // 16 jobs = 4 signals x 2 rowgroups x 2 colgroups
#pragma unroll
  for (int rep = 0; rep < 2; ++rep) {
    const int job = wu + rep * 8;
    const int s  = job & 3;
    const int r0 = ((job >> 2) & 1) * 16;
    const int c0 = ((job >> 3) & 1) * 16;
    v8f d = {0.f, 0.f, 0.f, 0.f, 0.f, 0.f, 0.f, 0.f};
#pragma unroll
    for (int q = 0; q < 7; ++q) {
      const int j0 = 4 * q + 2 * lhi;
      // two consecutive rows, same column -> contiguous in transposed buffer
      v2f bb = *(const v2f*)&sHT[s][c0 + lmod][r0 + j0];  // ds_load_b64
      d = __builtin_amdgcn_wmma_f32_16x16x4_f32(false, band[q], false, bb,
                                                (short)0, d, false, false);
    }
#pragma unroll
    for (int v = 0; v < 8; ++v)
      sO[s][r0 + 8 * lhi + v][c0 + lmod] = d[v];
  }
  __syncthreads();

  // ---- SSIM per valid pixel + block reduction ----
  float lsum = 0.0f;
#pragma unroll
  for (int i = tid; i < 32 * 32; i += 256) {
    int r = i >> 5, q = i & 31;
    int gr = tr0 + r, gc = tc0 + q;
    if (gr < OUTD && gc < OUTD) {
      float mu1 = sO[0][r][q];
      float epp = sO[1][r][q];
      float ept = sO[2][r][q];
      float mu2 = sO[3][r][q];
      float mu12 = mu1 * mu2;
      float mu1sq = mu1 * mu1;
      float mu2sq = mu2 * mu2;
      float var1 = epp - mu1sq;
      float var2 = mu2 - mu2sq;     // conv(t^2) == conv(t) == mu2
      float covar = ept - mu12;
      const float k1 = 1e-4f;       // (0.01)^2
      const float k2 = 9e-4f;       // (0.03)^2
      float cs = (2.0f * covar + k2) / (var1 + var2 + k2);
      float ss = (2.0f * mu12 + k1) / (mu1sq + mu2sq + k1) * cs;
      lsum += ss;
    }
  }
  sRed[tid] = lsum;
  __syncthreads();
  for (int off = 128; off > 0; off >>= 1) {
    if (tid < off) sRed[tid] += sRed[tid + off];
    __syncthreads();
  }
  if (tid == 0) atomicAdd(&acc[img], sRed[0]);
}

// ---------------------------------------------------------------------------
// Kernel 3: loss = 1 - mean_{b,c}( relu( sum_bc / 502^2 ) )
// ---------------------------------------------------------------------------
__global__ void finalize_k(const float* __restrict__ acc,
                           float* __restrict__ out) {
  __shared__ float sr[256];
  int tid = threadIdx.x;
  float v = 0.0f;
  if (tid < NIMG) {
    float m = acc[tid] * (1.0f / ((float)OUTD * (float)OUTD));
    v = m > 0.0f ? m : 0.0f;
  }
  sr[tid] = v;
  __syncthreads();
  for (int off = 128; off > 0; off >>= 1) {
    if (tid < off) sr[tid] += sr[tid + off];
    __syncthreads();
  }
  if (tid == 0) out[0] = 1.0f - sr[0] / (float)NIMG;
}

// ---------------------------------------------------------------------------
extern "C" void kernel_launch(void* const* d_in, const int* in_sizes, int n_in,
                              void* d_out, int out_size, void* d_ws,
                              size_t ws_size, hipStream_t stream) {
  const float* pred   = (const float*)d_in[0];
  const int*   target = (const int*)d_in[1];
  float* out = (float*)d_out;

  // workspace layout: [ p : B*C*H*W f32 ][ acc : 152 f32 ]
  float* ws_p = (float*)d_ws;
  float* accv = ws_p + (size_t)BB * CC * HWc;

  zero_k<<<1, 256, 0, stream>>>(accv);

  int npix = BB * HWc;
  softmax_k<<<(npix + 255) / 256, 256, 0, stream>>>(pred, ws_p);

  ssim_conv_k<<<dim3(256, NIMG), 256, 0, stream>>>(ws_p, target, accv);

  finalize_k<<<1, 256, 0, stream>>>(accv, out);
}